// IPSNet_78434692759623
// MI455X (gfx1250) — compile-verified
//
#include <hip/hip_runtime.h>
#include <math.h>
#include <stdint.h>

// ---------------- problem constants (reference defaults) ----------------
#define B_    8
#define N1_   33792
#define D_    128
#define H_    8
#define DK_   64
#define M_    1024
#define S_    1024
#define N2_   1024          // emb2 / attn2 length
#define REST_ 32768         // N1_ - M_
#define NTOT_ 33792         // REST_ + N2_
#define P_    65536         // pow2 padding for bitonic sort
#define NTILE_ (N1_ / 16)   // 2112 WMMA row-tiles per batch
#define LTILE_ 4096         // LDS sort tile (32 KB of u64)
#define TPAD_  (16 * 132)   // padded LDS floats per score tile

typedef __attribute__((ext_vector_type(2))) float v2f;
typedef __attribute__((ext_vector_type(8))) float v8f;
typedef unsigned long long u64;
typedef unsigned int       u32;

// ---------------- workspace layout (bytes) ----------------
constexpr size_t OFF_WQ  = 0;                          // 128*16*4      = 8192
constexpr size_t OFF_A1  = 8192;                       // 64*33792*4    = 8650752
constexpr size_t OFF_MX1 = OFF_A1  + 8650752;
constexpr size_t OFF_Z1  = OFF_MX1 + 256;
constexpr size_t OFF_S1  = OFF_Z1  + 256;              // 8*33792*4     = 1081344
constexpr size_t OFF_K1  = OFF_S1  + 1081344;          // 8*65536*8     = 4194304
constexpr size_t OFF_MX2 = OFF_K1  + 4194304;
constexpr size_t OFF_Z2  = OFF_MX2 + 256;
constexpr size_t OFF_S2  = OFF_Z2  + 256;
constexpr size_t OFF_K2  = OFF_S2  + 1081344;          // + 4194304 -> ~19.2 MB

// ---------------- output layout (floats, concat in return order) -------
constexpr size_t OUT_MEMEMB  = 0;                      // (8,1024,128)
constexpr size_t OUT_MEMIDX  = 1048576;                // (8,1024)
constexpr size_t OUT_SUBEMB  = 1056768;                // (8,1024,128)
constexpr size_t OUT_SUBATTN = 2105344;                // (8,8,1,1024)

// ---------------- helpers ----------------
__device__ __forceinline__ u32 f2ord(float f) {        // monotone float->uint
  u32 u = __float_as_uint(f);
  return (u & 0x80000000u) ? ~u : (u | 0x80000000u);
}
__device__ __forceinline__ u32 topIdx(const u64* keys, int b, int i) {
  return (u32)keys[(size_t)b * P_ + i];                // low 32 bits = index
}

// ======================================================================
// 1) wq_pad[k][n] = 0.125 * sum_kk w_k[k,n,kk] * q[n,kk]   (n<8; else 0)
// ======================================================================
__global__ __launch_bounds__(256) void k_wq(const float* __restrict__ w_k,
                                            const float* __restrict__ q,
                                            float* __restrict__ wq) {
  int t = blockIdx.x * 256 + threadIdx.x;              // 0 .. 2047
  if (t >= 128 * 16) return;
  int k = t >> 4, n = t & 15;
  float acc = 0.f;
  if (n < H_) {
    const float* wrow = w_k + (size_t)k * (H_ * DK_) + (size_t)n * DK_;
    const float* qrow = q + (size_t)n * DK_;
    for (int kk = 0; kk < DK_; ++kk) acc += wrow[kk] * qrow[kk];
    acc *= 0.125f;                                     // 1/sqrt(64), exact
  }
  wq[t] = acc;
}

// ======================================================================
// 2) attn1[b,h,n] via V_WMMA_F32_16X16X4_F32.
//    8 waves/block, one 16-row emb1 tile per wave.
//    A-tile staged with GLOBAL_LOAD_ASYNC_TO_LDS_B128 (ASYNCcnt path);
//    B fragments preloaded to registers so the MMA loop has no VMEM stalls.
// ======================================================================
__global__ __launch_bounds__(256) void k_scores(const float* __restrict__ emb1,
                                                const float* __restrict__ wq,
                                                float* __restrict__ attn1) {
  __shared__ __align__(16) float tileA[8 * TPAD_];     // 67.6 KB, per-wave slices
  int w    = threadIdx.x >> 5;                         // wave in block
  int lane = threadIdx.x & 31;
  int tile = blockIdx.x * 8 + w;                       // 0 .. B_*NTILE_-1
  int b    = tile / NTILE_;
  int n0   = (tile % NTILE_) * 16;

  float* tileW = &tileA[w * TPAD_];
  const float* src = emb1 + ((size_t)b * N1_ + n0) * D_;
  if (tile + 8 < B_ * NTILE_) __builtin_prefetch(src + 8 * 16 * D_, 0, 1);

  // async global -> LDS staging: 16 rows x 128 f32 (8 KB) per wave,
  // b128 per lane per step, no VGPR round-trip.
  unsigned ldsBase = (unsigned)(uintptr_t)tileW;       // low 32 bits = LDS offset
  #pragma unroll
  for (int i = 0; i < 16; ++i) {
    int linear = (i * 32 + lane) * 4;
    int rr = linear >> 7, cc = linear & 127;
    unsigned lds_addr = ldsBase + (unsigned)((rr * 132 + cc) * 4);
    unsigned long long gaddr = (unsigned long long)(uintptr_t)(src + rr * D_ + cc);
    asm volatile("global_load_async_to_lds_b128 %0, %1, off"
                 :: "v"(lds_addr), "v"(gaddr) : "memory");
  }

  // B fragments (loop-invariant per lane): preload while staging is in flight
  int r     = lane & 15;                               // row (A) / column N (B,C)
  int khalf = (lane >> 4) * 2;
  float bqa[32], bqb[32];
  #pragma unroll
  for (int t = 0; t < 32; ++t) {
    int ka = 4 * t + khalf;
    bqa[t] = wq[ka * 16 + r];
    bqb[t] = wq[(ka + 1) * 16 + r];
  }

  asm volatile("s_wait_asynccnt 0" ::: "memory");      // LDS tile ready (per-wave)

  // ISA A-layout (16x4 f32): lanes 0-15 rows, VGPR0/1 = K,K+1; lanes 16-31 = K+2,K+3
  v8f acc = {};
  #pragma unroll
  for (int t = 0; t < 32; ++t) {
    int ka = 4 * t + khalf;
    v2f a, bb;
    a.x  = tileW[r * 132 + ka];
    a.y  = tileW[r * 132 + ka + 1];
    bb.x = bqa[t];
    bb.y = bqb[t];
    acc = __builtin_amdgcn_wmma_f32_16x16x4_f32(false, a, false, bb,
                                                (short)0, acc, false, false);
  }

  // C-layout: VGPR i -> row n0+i (lanes 0-15) / n0+8+i (lanes 16-31), col = lane&15
  int h = r;
  if (h < H_) {
    int rowbase = n0 + ((lane >> 4) ? 8 : 0);
    float* dst = attn1 + ((size_t)(b * H_ + h)) * N1_ + rowbase;
    float4 lo = { acc[0], acc[1], acc[2], acc[3] };
    float4 hi = { acc[4], acc[5], acc[6], acc[7] };
    *reinterpret_cast<float4*>(dst)     = lo;
    *reinterpret_cast<float4*>(dst + 4) = hi;
  }
}

// ======================================================================
// 3) per-(b,h) softmax stats over attn1 rows
// ======================================================================
__global__ __launch_bounds__(256) void k_rowstat1(const float* __restrict__ attn1,
                                                  float* __restrict__ mx,
                                                  float* __restrict__ Z) {
  int bh = blockIdx.x;
  const float* row = attn1 + (size_t)bh * N1_;
  __shared__ float sh[256];
  float m = -INFINITY;
  for (int n = threadIdx.x; n < N1_; n += 256) m = fmaxf(m, row[n]);
  sh[threadIdx.x] = m; __syncthreads();
  for (int s = 128; s > 0; s >>= 1) {
    if (threadIdx.x < s) sh[threadIdx.x] = fmaxf(sh[threadIdx.x], sh[threadIdx.x + s]);
    __syncthreads();
  }
  float mm = sh[0]; __syncthreads();
  float sum = 0.f;
  for (int n = threadIdx.x; n < N1_; n += 256) sum += expf(row[n] - mm);
  sh[threadIdx.x] = sum; __syncthreads();
  for (int s = 128; s > 0; s >>= 1) {
    if (threadIdx.x < s) sh[threadIdx.x] += sh[threadIdx.x + s];
    __syncthreads();
  }
  if (threadIdx.x == 0) { mx[bh] = mm; Z[bh] = sh[0]; }
}

// ======================================================================
// 4) soft1[b,n] = (1/8) sum_h exp(attn1-mx)/Z
// ======================================================================
__global__ __launch_bounds__(256) void k_soft1(const float* __restrict__ attn1,
                                               const float* __restrict__ mx,
                                               const float* __restrict__ Z,
                                               float* __restrict__ soft1) {
  int t = blockIdx.x * 256 + threadIdx.x;
  if (t >= B_ * N1_) return;
  int b = t / N1_, n = t % N1_;
  float s = 0.f;
  for (int h = 0; h < H_; ++h) {
    int bh = b * H_ + h;
    s += expf(attn1[(size_t)bh * N1_ + n] - mx[bh]) / Z[bh];
  }
  soft1[t] = s * 0.125f;
}

// ======================================================================
// 5) key build: (~ord(value))<<32 | index  -> ascending sort gives
//    value-descending, index-ascending (lax.top_k tie rule). Pads last.
// ======================================================================
__global__ __launch_bounds__(256) void k_fillkeys(u64* __restrict__ keys,
                                                  const float* __restrict__ vals,
                                                  int n_valid, int stride) {
  int t = blockIdx.x * 256 + threadIdx.x;              // 0 .. B_*P_-1
  int b = t >> 16, i = t & (P_ - 1);
  u64 key;
  if (i < n_valid) {
    u32 kv = ~f2ord(vals[(size_t)b * stride + i]);
    key = ((u64)kv << 32) | (u32)i;
  } else {
    key = ~0ull;
  }
  keys[t] = key;
}

// ======================================================================
// 6a) LDS-resident full bitonic sort of 4096-key tiles (stages k=2..4096).
//     Direction from batch-local global index -> tiles come out alternately
//     asc/desc exactly as the k=8192 merge expects.
// ======================================================================
__global__ __launch_bounds__(256) void k_sort_local(u64* __restrict__ keys) {
  __shared__ __align__(16) u64 sh[LTILE_];
  int gbase = blockIdx.x * LTILE_;                     // memory base
  int lbase = gbase & (P_ - 1);                        // batch-local index base
  for (int i = threadIdx.x; i < LTILE_; i += 256) sh[i] = keys[gbase + i];
  __syncthreads();
  for (int k = 2; k <= LTILE_; k <<= 1) {
    for (int j = k >> 1; j > 0; j >>= 1) {
      for (int t = threadIdx.x; t < LTILE_ / 2; t += 256) {
        int i = ((t & ~(j - 1)) << 1) | (t & (j - 1)); // bit log2(j) clear
        int l = i | j;
        bool asc = (((lbase + i) & k) == 0);
        u64 a = sh[i], c = sh[l];
        if (asc ? (a > c) : (a < c)) { sh[i] = c; sh[l] = a; }
      }
      __syncthreads();
    }
  }
  for (int i = threadIdx.x; i < LTILE_; i += 256) keys[gbase + i] = sh[i];
}

// ======================================================================
// 6b) LDS-resident merge tail for stage k: steps j=2048..1 (direction is
//     uniform per tile for k >= 8192).
// ======================================================================
__global__ __launch_bounds__(256) void k_merge_local(u64* __restrict__ keys, int k) {
  __shared__ __align__(16) u64 sh[LTILE_];
  int gbase = blockIdx.x * LTILE_;
  int lbase = gbase & (P_ - 1);
  bool asc = ((lbase & k) == 0);
  for (int i = threadIdx.x; i < LTILE_; i += 256) sh[i] = keys[gbase + i];
  __syncthreads();
  for (int j = LTILE_ / 2; j > 0; j >>= 1) {
    for (int t = threadIdx.x; t < LTILE_ / 2; t += 256) {
      int i = ((t & ~(j - 1)) << 1) | (t & (j - 1));
      int l = i | j;
      u64 a = sh[i], c = sh[l];
      if (asc ? (a > c) : (a < c)) { sh[i] = c; sh[l] = a; }
    }
    __syncthreads();
  }
  for (int i = threadIdx.x; i < LTILE_; i += 256) keys[gbase + i] = sh[i];
}

// ======================================================================
// 6c) global compare-exchange step (only needed for j >= 4096)
// ======================================================================
__global__ __launch_bounds__(256) void k_bitonic(u64* __restrict__ keys, int j, int k) {
  int t = blockIdx.x * 256 + threadIdx.x;              // 0 .. B_*P_-1
  int i = t & (P_ - 1);
  int base = t & ~(P_ - 1);
  int ixj = i ^ j;
  if (ixj > i) {
    u64 a = keys[base + i], c = keys[base + ixj];
    bool asc = (i & k) == 0;
    if (asc ? (a > c) : (a < c)) {
      keys[base + i]   = c;
      keys[base + ixj] = a;
    }
  }
}

// ======================================================================
// 7) mem_emb + mem_idx from top_idx1[:, :M]
// ======================================================================
__global__ __launch_bounds__(256) void k_mem_out(const float* __restrict__ emb1,
                                                 const int* __restrict__ idx1,
                                                 const u64* __restrict__ keys1,
                                                 float* __restrict__ mem_emb,
                                                 float* __restrict__ mem_idx) {
  int t = blockIdx.x * 256 + threadIdx.x;              // B_*M_*D_
  if (t >= B_ * M_ * D_) return;
  int b = t / (M_ * D_);
  int rem = t % (M_ * D_);
  int m = rem / D_, d = rem % D_;
  u32 idx = topIdx(keys1, b, m);
  mem_emb[t] = emb1[((size_t)b * N1_ + idx) * D_ + d];
  if (d == 0) mem_idx[(size_t)b * M_ + m] = (float)idx1[(size_t)b * N1_ + idx];
}

// ======================================================================
// 8) softmax stats over virtual concat [attn1 gathered at rest | attn2]
// ======================================================================
__global__ __launch_bounds__(256) void k_rowstat2(const float* __restrict__ attn1,
                                                  const float* __restrict__ attn2,
                                                  const u64* __restrict__ keys1,
                                                  float* __restrict__ mx,
                                                  float* __restrict__ Z) {
  int bh = blockIdx.x;
  int b = bh >> 3;
  const float* row1 = attn1 + (size_t)bh * N1_;
  const float* row2 = attn2 + (size_t)bh * N2_;
  __shared__ float sh[256];
  float m = -INFINITY;
  for (int j = threadIdx.x; j < NTOT_; j += 256) {
    float v = (j < REST_) ? row1[topIdx(keys1, b, M_ + j)] : row2[j - REST_];
    m = fmaxf(m, v);
  }
  sh[threadIdx.x] = m; __syncthreads();
  for (int s = 128; s > 0; s >>= 1) {
    if (threadIdx.x < s) sh[threadIdx.x] = fmaxf(sh[threadIdx.x], sh[threadIdx.x + s]);
    __syncthreads();
  }
  float mm = sh[0]; __syncthreads();
  float sum = 0.f;
  for (int j = threadIdx.x; j < NTOT_; j += 256) {
    float v = (j < REST_) ? row1[topIdx(keys1, b, M_ + j)] : row2[j - REST_];
    sum += expf(v - mm);
  }
  sh[threadIdx.x] = sum; __syncthreads();
  for (int s = 128; s > 0; s >>= 1) {
    if (threadIdx.x < s) sh[threadIdx.x] += sh[threadIdx.x + s];
    __syncthreads();
  }
  if (threadIdx.x == 0) { mx[bh] = mm; Z[bh] = sh[0]; }
}

// ======================================================================
// 9) soft2[b,j] over the virtual concat
// ======================================================================
__global__ __launch_bounds__(256) void k_soft2(const float* __restrict__ attn1,
                                               const float* __restrict__ attn2,
                                               const u64* __restrict__ keys1,
                                               const float* __restrict__ mx,
                                               const float* __restrict__ Z,
                                               float* __restrict__ soft2) {
  int t = blockIdx.x * 256 + threadIdx.x;
  if (t >= B_ * NTOT_) return;
  int b = t / NTOT_, j = t % NTOT_;
  float s = 0.f;
  if (j < REST_) {
    u32 src = topIdx(keys1, b, M_ + j);
    for (int h = 0; h < H_; ++h) {
      int bh = b * H_ + h;
      s += expf(attn1[(size_t)bh * N1_ + src] - mx[bh]) / Z[bh];
    }
  } else {
    int s2 = j - REST_;
    for (int h = 0; h < H_; ++h) {
      int bh = b * H_ + h;
      s += expf(attn2[(size_t)bh * N2_ + s2] - mx[bh]) / Z[bh];
    }
  }
  soft2[t] = s * 0.125f;
}

// ======================================================================
// 10) sub_emb: compose gathers top_idx2 -> (rest|emb2)
// ======================================================================
__global__ __launch_bounds__(256) void k_sub_emb(const float* __restrict__ emb1,
                                                 const float* __restrict__ emb2,
                                                 const u64* __restrict__ keys1,
                                                 const u64* __restrict__ keys2,
                                                 float* __restrict__ sub_emb) {
  int t = blockIdx.x * 256 + threadIdx.x;              // B_*S_*D_
  if (t >= B_ * S_ * D_) return;
  int b = t / (S_ * D_);
  int rem = t % (S_ * D_);
  int s = rem / D_, d = rem % D_;
  u32 j = topIdx(keys2, b, s);
  float v;
  if (j < REST_) {
    u32 src = topIdx(keys1, b, M_ + j);
    v = emb1[((size_t)b * N1_ + src) * D_ + d];
  } else {
    v = emb2[((size_t)b * N2_ + (j - REST_)) * D_ + d];
  }
  sub_emb[t] = v;
}

// ======================================================================
// 11) sub_attn
// ======================================================================
__global__ __launch_bounds__(256) void k_sub_attn(const float* __restrict__ attn1,
                                                  const float* __restrict__ attn2,
                                                  const u64* __restrict__ keys1,
                                                  const u64* __restrict__ keys2,
                                                  float* __restrict__ sub_attn) {
  int t = blockIdx.x * 256 + threadIdx.x;              // B_*H_*S_
  if (t >= B_ * H_ * S_) return;
  int b = t / (H_ * S_);
  int h = (t / S_) % H_;
  int s = t % S_;
  u32 j = topIdx(keys2, b, s);
  int bh = b * H_ + h;
  float v = (j < REST_) ? attn1[(size_t)bh * N1_ + topIdx(keys1, b, M_ + j)]
                        : attn2[(size_t)bh * N2_ + (j - REST_)];
  sub_attn[t] = v;
}

// ======================================================================
extern "C" void kernel_launch(void* const* d_in, const int* in_sizes, int n_in,
                              void* d_out, int out_size, void* d_ws, size_t ws_size,
                              hipStream_t stream) {
  (void)in_sizes; (void)n_in; (void)out_size; (void)ws_size;
  const float* emb1  = (const float*)d_in[0];
  const int*   idx1  = (const int*)  d_in[1];
  const float* emb2  = (const float*)d_in[2];
  const float* attn2 = (const float*)d_in[3];
  const float* w_k   = (const float*)d_in[4];
  const float* q     = (const float*)d_in[5];
  float* out = (float*)d_out;
  char*  ws  = (char*)d_ws;

  float* wq    = (float*)(ws + OFF_WQ);
  float* attn1 = (float*)(ws + OFF_A1);
  float* mx1   = (float*)(ws + OFF_MX1);
  float* Z1    = (float*)(ws + OFF_Z1);
  float* soft1 = (float*)(ws + OFF_S1);
  u64*   keys1 = (u64*)  (ws + OFF_K1);
  float* mx2   = (float*)(ws + OFF_MX2);
  float* Z2    = (float*)(ws + OFF_Z2);
  float* soft2 = (float*)(ws + OFF_S2);
  u64*   keys2 = (u64*)  (ws + OFF_K2);

  auto sort_keys = [&](u64* keys) {
    // stages k=2..4096 entirely in LDS (tiles alternate asc/desc correctly)
    k_sort_local<<<(B_ * P_) / LTILE_, 256, 0, stream>>>(keys);
    for (int k = 8192; k <= P_; k <<= 1) {
      for (int j = k >> 1; j >= LTILE_; j >>= 1)        // cross-tile steps
        k_bitonic<<<(B_ * P_) / 256, 256, 0, stream>>>(keys, j, k);
      k_merge_local<<<(B_ * P_) / LTILE_, 256, 0, stream>>>(keys, k);
    }
  };

  // 1) projection fold: wq = (w_k @ q) * 1/sqrt(Dk), padded to 128x16
  k_wq<<<8, 256, 0, stream>>>(w_k, q, wq);

  // 2) scores via f32 WMMA + async-to-LDS staging
  k_scores<<<(B_ * NTILE_) / 8, 256, 0, stream>>>(emb1, wq, attn1);

  // 3-4) soft1
  k_rowstat1<<<B_ * H_, 256, 0, stream>>>(attn1, mx1, Z1);
  k_soft1<<<(B_ * N1_ + 255) / 256, 256, 0, stream>>>(attn1, mx1, Z1, soft1);

  // 5-6) full descending argsort of soft1
  k_fillkeys<<<(B_ * P_) / 256, 256, 0, stream>>>(keys1, soft1, N1_, N1_);
  sort_keys(keys1);

  // 7) mem_emb / mem_idx
  k_mem_out<<<(B_ * M_ * D_) / 256, 256, 0, stream>>>(
      emb1, idx1, keys1, out + OUT_MEMEMB, out + OUT_MEMIDX);

  // 8-9) soft2 over virtual concat
  k_rowstat2<<<B_ * H_, 256, 0, stream>>>(attn1, attn2, keys1, mx2, Z2);
  k_soft2<<<(B_ * NTOT_ + 255) / 256, 256, 0, stream>>>(attn1, attn2, keys1, mx2, Z2, soft2);

  // 10-11) top-1024 of soft2 (full sort; only first S entries consumed)
  k_fillkeys<<<(B_ * P_) / 256, 256, 0, stream>>>(keys2, soft2, NTOT_, NTOT_);
  sort_keys(keys2);

  // 12) sub_emb / sub_attn via composed gathers (no materialized concat)
  k_sub_emb<<<(B_ * S_ * D_) / 256, 256, 0, stream>>>(
      emb1, emb2, keys1, keys2, out + OUT_SUBEMB);
  k_sub_attn<<<(B_ * H_ * S_) / 256, 256, 0, stream>>>(
      attn1, attn2, keys1, keys2, out + OUT_SUBATTN);
}